// BalancedGAT_53412213293198
// MI455X (gfx1250) — compile-verified
//
#include <hip/hip_runtime.h>
#include <math.h>

#define NN   50000      // nodes
#define EE   600000     // edges (without self loops)
#define ET   (EE + NN)  // edges + self loops
#define OUTC 40
#define CDIV(a,b) (((a)+(b)-1)/(b))

typedef float v2f __attribute__((ext_vector_type(2)));
typedef float v8f __attribute__((ext_vector_type(8)));

// ---------------------------------------------------------------------------
// f32 WMMA GEMM with LDS-staged A:
//   Hout[NN x Mo] = A[NN x K] @ W[K x Mo]
// One block (8 waves) covers an RT*16-row stripe; the stripe of A is one
// contiguous memory range -> coalesced float4 copy into LDS, then each wave
// computes a 16x16 tile with V_WMMA_F32_16X16X4_F32, A fragments from LDS.
// Ragged Mo handled by clamping the load column (garbage columns are never
// stored, so no zeroing / exec masking needed inside the K loop).
// ---------------------------------------------------------------------------
template<int RT>
__global__ __launch_bounds__(256) void gemm_wmma_lds(
    const float* __restrict__ A, const float* __restrict__ W,
    float* __restrict__ Hout, int K, int Mo, int mtiles)
{
    __shared__ float sA[RT * 16 * 128];          // K <= 128
    const int tid  = threadIdx.x;
    const int row0 = blockIdx.x * (RT * 16);
    const int nrows = min(RT * 16, NN - row0);

    // cooperative, fully coalesced copy of the contiguous A stripe into LDS
    {
        const float4* __restrict__ g4 = (const float4*)(A + (size_t)row0 * K);
        float4* s4 = (float4*)sA;
        const int n4 = (nrows * K) >> 2;
        for (int i = tid; i < n4; i += 256) s4[i] = g4[i];
    }
    __syncthreads();

    const int wave = tid >> 5;
    const int lane = tid & 31;
    const int rloc = wave / mtiles;              // wave-uniform
    const int colt = wave % mtiles;
    if (rloc >= RT) return;                      // uniform exit, EXEC stays full
    const int row = row0 + rloc * 16;
    if (row >= NN) return;                       // uniform exit (ragged last block)

    const int half = lane >> 4;                  // 0 -> K{0,1}, 1 -> K{2,3}
    const int l15  = lane & 15;
    const int col  = colt * 16 + l15;
    const int colc = col < Mo ? col : Mo - 1;    // clamp keeps loads in bounds

    v8f c = {0.f,0.f,0.f,0.f,0.f,0.f,0.f,0.f};
    const float* sArow = sA + (rloc * 16 + l15) * K;

#pragma unroll 8
    for (int k = 0; k < K; k += 4) {
        const int kb = k + half * 2;
        const v2f a = *(const v2f*)(sArow + kb); // 8B aligned -> ds_load_b64
        v2f b;
        b.x = W[(size_t)kb       * Mo + colc];
        b.y = W[(size_t)(kb + 1) * Mo + colc];
        c = __builtin_amdgcn_wmma_f32_16x16x4_f32(
                /*neg_a=*/false, a, /*neg_b=*/false, b,
                /*c_mod=*/(short)0, c, /*reuse_a=*/false, /*reuse_b=*/false);
    }
    if (col < Mo) {
        const int mbase = row + half * 8;
#pragma unroll
        for (int r = 0; r < 8; ++r)
            Hout[(size_t)(mbase + r) * Mo + col] = c[r];
    }
}

// ---------------------------------------------------------------------------
// attention scores: s[n,h] = <h[n,h,:], a_s[h,:]>,  d likewise
// ---------------------------------------------------------------------------
template<int H, int C>
__global__ void scores_kernel(const float* __restrict__ hbuf,
                              const float* __restrict__ a_s,
                              const float* __restrict__ a_d,
                              float* __restrict__ s, float* __restrict__ d)
{
    int i = blockIdx.x * blockDim.x + threadIdx.x;
    if (i >= NN * H) return;
    const int n = i / H, hh = i % H;
    const float* hp = hbuf + (size_t)n * (H * C) + hh * C;
    float ss = 0.f, dd = 0.f;
    for (int c = 0; c < C; ++c) {
        const float v = hp[c];
        ss += v * a_s[hh * C + c];
        dd += v * a_d[hh * C + c];
    }
    s[i] = ss; d[i] = dd;
}

__global__ void fill_kernel(float* __restrict__ p, float v, int n)
{
    int i = blockIdx.x * blockDim.x + threadIdx.x;
    if (i < n) p[i] = v;
}

__device__ inline void atomicMaxFloat(float* addr, float val)
{
    if (val >= 0.f) atomicMax((int*)addr, __float_as_int(val));
    else            atomicMin((unsigned int*)addr, (unsigned int)__float_as_int(val));
}

__device__ inline void edge_uv(int e, const int* __restrict__ src,
                               const int* __restrict__ dst, int& u, int& v)
{
    if (e < EE) { u = src[e]; v = dst[e]; }
    else        { u = e - EE; v = e - EE; }   // implicit self loop
}

__device__ inline float lrelu(float x) { return x > 0.f ? x : 0.2f * x; }

template<int H>
__global__ void edge_max_kernel(const int* __restrict__ src, const int* __restrict__ dst,
                                const float* __restrict__ s, const float* __restrict__ d,
                                float* __restrict__ nmax)
{
    int i = blockIdx.x * blockDim.x + threadIdx.x;
    if (i >= ET * H) return;
    const int e = i / H, hh = i % H;
    int u, v; edge_uv(e, src, dst, u, v);
    const float ev = lrelu(s[u * H + hh] + d[v * H + hh]);
    atomicMaxFloat(&nmax[v * H + hh], ev);
}

template<int H>
__global__ void edge_den_kernel(const int* __restrict__ src, const int* __restrict__ dst,
                                const float* __restrict__ s, const float* __restrict__ d,
                                const float* __restrict__ nmax, float* __restrict__ nden)
{
    int i = blockIdx.x * blockDim.x + threadIdx.x;
    if (i >= ET * H) return;
    const int e = i / H, hh = i % H;
    int u, v; edge_uv(e, src, dst, u, v);
    const float ev = lrelu(s[u * H + hh] + d[v * H + hh]);
    atomicAdd(&nden[v * H + hh], __expf(ev - nmax[v * H + hh]));
}

template<int H>
__global__ void edge_agg_kernel(const int* __restrict__ src, const int* __restrict__ dst,
                                const float* __restrict__ s, const float* __restrict__ d,
                                const float* __restrict__ nmax, const float* __restrict__ nden,
                                const float* __restrict__ hbuf, float* __restrict__ agg, int C)
{
    int i = blockIdx.x * blockDim.x + threadIdx.x;
    if (i >= ET * H) return;
    const int e = i / H, hh = i % H;
    int u, v; edge_uv(e, src, dst, u, v);
    const float ev    = lrelu(s[u * H + hh] + d[v * H + hh]);
    const float alpha = __expf(ev - nmax[v * H + hh]) / (nden[v * H + hh] + 1e-16f);
    const float* hs = hbuf + (size_t)(u * H + hh) * C;
    float*       op = agg  + (size_t)(v * H + hh) * C;
    for (int c = 0; c < C; ++c)
        atomicAdd(&op[c], hs[c] * alpha);
}

// ---------------------------------------------------------------------------
// BatchNorm statistics: one block per column (biased variance, as in jnp.var)
// ---------------------------------------------------------------------------
__global__ __launch_bounds__(256) void bn_stats_kernel(
    const float* __restrict__ x, int D,
    float* __restrict__ mean, float* __restrict__ rstd)
{
    const int col = blockIdx.x;
    float s = 0.f, s2 = 0.f;
    for (int r = threadIdx.x; r < NN; r += blockDim.x) {
        const float v = x[(size_t)r * D + col];
        s += v; s2 += v * v;
    }
    __shared__ float sh[256], sh2[256];
    sh[threadIdx.x] = s; sh2[threadIdx.x] = s2;
    __syncthreads();
    for (int o = 128; o > 0; o >>= 1) {
        if (threadIdx.x < o) {
            sh[threadIdx.x]  += sh[threadIdx.x + o];
            sh2[threadIdx.x] += sh2[threadIdx.x + o];
        }
        __syncthreads();
    }
    if (threadIdx.x == 0) {
        const float mu  = sh[0] / (float)NN;
        const float var = sh2[0] / (float)NN - mu * mu;
        mean[col] = mu;
        rstd[col] = rsqrtf(var + 1e-5f);
    }
}

// y = [resin +] (elu?)(g*(x-mu)*rstd + be); optional second copy (residual save)
__global__ void bn_apply_kernel(const float* __restrict__ x,
                                const float* __restrict__ mean, const float* __restrict__ rstd,
                                const float* __restrict__ g, const float* __restrict__ be,
                                const float* __restrict__ resin,
                                float* __restrict__ out, float* __restrict__ resout,
                                int D, int do_elu)
{
    int i = blockIdx.x * blockDim.x + threadIdx.x;
    if (i >= NN * D) return;
    const int col = i % D;
    float v = g[col] * (x[i] - mean[col]) * rstd[col] + be[col];
    if (do_elu) v = v > 0.f ? v : expm1f(v);
    if (resin)  v += resin[i];
    out[i] = v;
    if (resout) resout[i] = v;
}

__global__ void logsoftmax_kernel(const float* __restrict__ x, float* __restrict__ out)
{
    int n = blockIdx.x * blockDim.x + threadIdx.x;
    if (n >= NN) return;
    const float* r = x + (size_t)n * OUTC;
    float m = -INFINITY;
    for (int j = 0; j < OUTC; ++j) m = fmaxf(m, r[j]);
    float sum = 0.f;
    for (int j = 0; j < OUTC; ++j) sum += __expf(r[j] - m);
    const float lse = m + logf(sum);
    for (int j = 0; j < OUTC; ++j) out[(size_t)n * OUTC + j] = r[j] - lse;
}

// ---------------------------------------------------------------------------
extern "C" void kernel_launch(void* const* d_in, const int* in_sizes, int n_in,
                              void* d_out, int out_size, void* d_ws, size_t ws_size,
                              hipStream_t stream)
{
    (void)in_sizes; (void)n_in; (void)out_size; (void)ws_size;

    const float* x   = (const float*)d_in[0];
    const int*   ei  = (const int*)d_in[1];
    const int*   src = ei;
    const int*   dst = ei + EE;
    const float* W1  = (const float*)d_in[2];
    const float* as1 = (const float*)d_in[3];
    const float* ad1 = (const float*)d_in[4];
    // biases (d_in[5], [11], [17], [23]) cancel inside BatchNorm -> skipped
    const float* g1  = (const float*)d_in[6];
    const float* be1 = (const float*)d_in[7];
    const float* Wm  = (const float*)d_in[8];
    const float* a_sm= (const float*)d_in[9];
    const float* a_dm= (const float*)d_in[10];
    const float* gm  = (const float*)d_in[12];
    const float* bem = (const float*)d_in[13];
    const float* W2  = (const float*)d_in[14];
    const float* as2 = (const float*)d_in[15];
    const float* ad2 = (const float*)d_in[16];
    const float* g2  = (const float*)d_in[18];
    const float* be2 = (const float*)d_in[19];
    const float* W3  = (const float*)d_in[20];
    const float* as3 = (const float*)d_in[21];
    const float* ad3 = (const float*)d_in[22];
    const float* g3  = (const float*)d_in[24];
    const float* be3 = (const float*)d_in[25];

    float* ws   = (float*)d_ws;
    float* hb   = ws;                        // N*128 projected features
    float* act  = hb   + (size_t)NN * 128;   // N*128 current activations
    float* res  = act  + (size_t)NN * 128;   // N*128 residual
    float* agg  = res  + (size_t)NN * 128;   // N*128 aggregation accumulator
    float* ssc  = agg  + (size_t)NN * 128;   // N*8 source scores
    float* dsc  = ssc  + (size_t)NN * 8;     // N*8 dest scores
    float* nmax = dsc  + (size_t)NN * 8;     // N*8 segment max
    float* nden = nmax + (size_t)NN * 8;     // N*8 segment denom
    float* mean = nden + (size_t)NN * 8;     // 128
    float* rstd = mean + 128;                // 128

    const int TB = 256;

    // ---------------- Layer 1: conv1  (128 -> 8 heads x 16, concat=128) -----
    {
        const int K = 128, Mo = 128, H = 8, C = 16, mt = Mo / 16;  // RT=1
        gemm_wmma_lds<1><<<CDIV(NN, 16), TB, 0, stream>>>(x, W1, hb, K, Mo, mt);
        scores_kernel<H, C><<<CDIV(NN * H, TB), TB, 0, stream>>>(hb, as1, ad1, ssc, dsc);
        fill_kernel<<<CDIV(NN * H, TB), TB, 0, stream>>>(nmax, -INFINITY, NN * H);
        fill_kernel<<<CDIV(NN * H, TB), TB, 0, stream>>>(nden, 0.f, NN * H);
        fill_kernel<<<CDIV(NN * Mo, TB), TB, 0, stream>>>(agg, 0.f, NN * Mo);
        edge_max_kernel<H><<<CDIV(ET * H, TB), TB, 0, stream>>>(src, dst, ssc, dsc, nmax);
        edge_den_kernel<H><<<CDIV(ET * H, TB), TB, 0, stream>>>(src, dst, ssc, dsc, nmax, nden);
        edge_agg_kernel<H><<<CDIV(ET * H, TB), TB, 0, stream>>>(src, dst, ssc, dsc, nmax, nden, hb, agg, C);
        bn_stats_kernel<<<Mo, TB, 0, stream>>>(agg, Mo, mean, rstd);
        bn_apply_kernel<<<CDIV(NN * Mo, TB), TB, 0, stream>>>(agg, mean, rstd, g1, be1,
                                                              nullptr, act, res, Mo, 1);
    }
    // ---------------- Layer mid: conv_mid (128 -> 128, 1 head, mean) + res --
    {
        const int K = 128, Mo = 128, H = 1, C = 128, mt = Mo / 16; // RT=1
        gemm_wmma_lds<1><<<CDIV(NN, 16), TB, 0, stream>>>(act, Wm, hb, K, Mo, mt);
        scores_kernel<H, C><<<CDIV(NN * H, TB), TB, 0, stream>>>(hb, a_sm, a_dm, ssc, dsc);
        fill_kernel<<<CDIV(NN * H, TB), TB, 0, stream>>>(nmax, -INFINITY, NN * H);
        fill_kernel<<<CDIV(NN * H, TB), TB, 0, stream>>>(nden, 0.f, NN * H);
        fill_kernel<<<CDIV(NN * Mo, TB), TB, 0, stream>>>(agg, 0.f, NN * Mo);
        edge_max_kernel<H><<<CDIV(ET * H, TB), TB, 0, stream>>>(src, dst, ssc, dsc, nmax);
        edge_den_kernel<H><<<CDIV(ET * H, TB), TB, 0, stream>>>(src, dst, ssc, dsc, nmax, nden);
        edge_agg_kernel<H><<<CDIV(ET * H, TB), TB, 0, stream>>>(src, dst, ssc, dsc, nmax, nden, hb, agg, C);
        bn_stats_kernel<<<Mo, TB, 0, stream>>>(agg, Mo, mean, rstd);
        bn_apply_kernel<<<CDIV(NN * Mo, TB), TB, 0, stream>>>(agg, mean, rstd, gm, bem,
                                                              res, act, nullptr, Mo, 1);
    }
    // ---------------- Layer 2: conv2 (128 -> 4 heads x 16, concat=64) -------
    {
        const int K = 128, Mo = 64, H = 4, C = 16, mt = Mo / 16;   // RT=2
        gemm_wmma_lds<2><<<CDIV(NN, 32), TB, 0, stream>>>(act, W2, hb, K, Mo, mt);
        scores_kernel<H, C><<<CDIV(NN * H, TB), TB, 0, stream>>>(hb, as2, ad2, ssc, dsc);
        fill_kernel<<<CDIV(NN * H, TB), TB, 0, stream>>>(nmax, -INFINITY, NN * H);
        fill_kernel<<<CDIV(NN * H, TB), TB, 0, stream>>>(nden, 0.f, NN * H);
        fill_kernel<<<CDIV(NN * Mo, TB), TB, 0, stream>>>(agg, 0.f, NN * Mo);
        edge_max_kernel<H><<<CDIV(ET * H, TB), TB, 0, stream>>>(src, dst, ssc, dsc, nmax);
        edge_den_kernel<H><<<CDIV(ET * H, TB), TB, 0, stream>>>(src, dst, ssc, dsc, nmax, nden);
        edge_agg_kernel<H><<<CDIV(ET * H, TB), TB, 0, stream>>>(src, dst, ssc, dsc, nmax, nden, hb, agg, C);
        bn_stats_kernel<<<Mo, TB, 0, stream>>>(agg, Mo, mean, rstd);
        bn_apply_kernel<<<CDIV(NN * Mo, TB), TB, 0, stream>>>(agg, mean, rstd, g2, be2,
                                                              nullptr, act, nullptr, Mo, 1);
    }
    // ---------------- Layer 3: conv3 (64 -> 40, 1 head, mean) + log_softmax -
    {
        const int K = 64, Mo = OUTC, H = 1, C = OUTC, mt = CDIV(Mo, 16); // RT=2
        gemm_wmma_lds<2><<<CDIV(NN, 32), TB, 0, stream>>>(act, W3, hb, K, Mo, mt);
        scores_kernel<H, C><<<CDIV(NN * H, TB), TB, 0, stream>>>(hb, as3, ad3, ssc, dsc);
        fill_kernel<<<CDIV(NN * H, TB), TB, 0, stream>>>(nmax, -INFINITY, NN * H);
        fill_kernel<<<CDIV(NN * H, TB), TB, 0, stream>>>(nden, 0.f, NN * H);
        fill_kernel<<<CDIV(NN * Mo, TB), TB, 0, stream>>>(agg, 0.f, NN * Mo);
        edge_max_kernel<H><<<CDIV(ET * H, TB), TB, 0, stream>>>(src, dst, ssc, dsc, nmax);
        edge_den_kernel<H><<<CDIV(ET * H, TB), TB, 0, stream>>>(src, dst, ssc, dsc, nmax, nden);
        edge_agg_kernel<H><<<CDIV(ET * H, TB), TB, 0, stream>>>(src, dst, ssc, dsc, nmax, nden, hb, agg, C);
        bn_stats_kernel<<<Mo, TB, 0, stream>>>(agg, Mo, mean, rstd);
        bn_apply_kernel<<<CDIV(NN * Mo, TB), TB, 0, stream>>>(agg, mean, rstd, g3, be3,
                                                              nullptr, hb, nullptr, Mo, 0);
        logsoftmax_kernel<<<CDIV(NN, TB), TB, 0, stream>>>(hb, (float*)d_out);
    }
}